// HeteroMessagePassingBank_7258494730302
// MI455X (gfx1250) — compile-verified
//
#include <hip/hip_runtime.h>
#include <hip/hip_bf16.h>

typedef float v2f __attribute__((ext_vector_type(2)));
typedef float v8f __attribute__((ext_vector_type(8)));

#define DIM 128

// ---------------------------------------------------------------------------
// Row-wise softmax: one wave32 per row, 4 floats per lane (float4 loads).
// ---------------------------------------------------------------------------
__global__ __launch_bounds__(256) void softmax_rows(const float* __restrict__ x,
                                                    float* __restrict__ p,
                                                    int n_rows) {
    const int lane = threadIdx.x & 31;
    const int wave = threadIdx.x >> 5;
    const int row  = blockIdx.x * 8 + wave;
    if (row >= n_rows) return;

    const float4 v = *(const float4*)(x + (size_t)row * DIM + lane * 4);
    float mx = fmaxf(fmaxf(v.x, v.y), fmaxf(v.z, v.w));
    #pragma unroll
    for (int off = 16; off >= 1; off >>= 1)
        mx = fmaxf(mx, __shfl_xor(mx, off, 32));

    const float e0 = __expf(v.x - mx), e1 = __expf(v.y - mx);
    const float e2 = __expf(v.z - mx), e3 = __expf(v.w - mx);
    float s = e0 + e1 + e2 + e3;
    #pragma unroll
    for (int off = 16; off >= 1; off >>= 1)
        s += __shfl_xor(s, off, 32);

    const float inv = 1.0f / s;
    float4 o; o.x = e0 * inv; o.y = e1 * inv; o.z = e2 * inv; o.w = e3 * inv;
    *(float4*)(p + (size_t)row * DIM + lane * 4) = o;
}

// ---------------------------------------------------------------------------
// W_eff = W_pos - softplus(W_neg_raw)   (numerically stable softplus)
// ---------------------------------------------------------------------------
__global__ void weight_eff(const float* __restrict__ wp,
                           const float* __restrict__ wn,
                           float* __restrict__ w, int n) {
    const int i = blockIdx.x * blockDim.x + threadIdx.x;
    if (i < n) {
        const float xv = wn[i];
        const float sp = fmaxf(xv, 0.0f) + log1pf(__expf(-fabsf(xv)));
        w[i] = wp[i] - sp;
    }
}

// ---------------------------------------------------------------------------
// Y[n_rows x 128] = P[n_rows x 128] @ W[128 x 128], fp32 WMMA 16x16x4.
// Block = 256 threads = 8 waves; each wave owns one 16x16 output tile
// (8 waves cover all 128 columns); block covers 16 rows.
// W (64 KB) is staged in LDS once per block.
// ---------------------------------------------------------------------------
__global__ __launch_bounds__(256) void gemm_wmma_f32(const float* __restrict__ P,
                                                     const float* __restrict__ W,
                                                     float* __restrict__ Y,
                                                     int n_rows) {
    __shared__ float Wlds[DIM * DIM];  // 64 KB of the 320 KB WGP LDS

    // Cooperative load of the whole 128x128 weight matrix into LDS.
    {
        const float4* Wg = (const float4*)W;
        float4*       Wl = (float4*)Wlds;
        #pragma unroll
        for (int i = 0; i < 16; ++i)               // 4096 float4 / 256 threads
            Wl[threadIdx.x + i * 256] = Wg[threadIdx.x + i * 256];
    }
    __syncthreads();

    const int lane  = threadIdx.x & 31;
    const int wave  = threadIdx.x >> 5;
    const int col0  = wave * 16;               // this wave's 16-col slab
    const int row0  = blockIdx.x * 16;         // this block's 16-row slab
    const int m     = lane & 15;               // row within tile (A), col (B & C)
    const int khalf = lane >> 4;               // 0: K+{0,1}, 1: K+{2,3}

    int arow = row0 + m;
    if (arow >= n_rows) arow = n_rows - 1;     // keep EXEC all-ones for WMMA
    const float* __restrict__ Ap = P + (size_t)arow * DIM;

    v8f c = {};

    #pragma unroll
    for (int k0 = 0; k0 < DIM; k0 += 4) {
        // A fragment (16x4 f32): lane l holds K = k0 + 2*khalf + {0,1}
        const float2 av = *(const float2*)(Ap + k0 + khalf * 2);
        v2f a; a[0] = av.x; a[1] = av.y;

        // B fragment (4x16 f32): VGPR r holds K = k0 + 2*khalf + r, N = lane&15
        v2f b;
        b[0] = Wlds[(k0 + 2 * khalf + 0) * DIM + col0 + m];
        b[1] = Wlds[(k0 + 2 * khalf + 1) * DIM + col0 + m];

        // D = A*B + C    (8 args: neg_a, A, neg_b, B, c_mod, C, reuse_a, reuse_b)
        c = __builtin_amdgcn_wmma_f32_16x16x4_f32(false, a, false, b,
                                                  (short)0, c, false, false);
    }

    // C/D layout: VGPR r -> row (r + 8*khalf), col (lane & 15)
    #pragma unroll
    for (int r = 0; r < 8; ++r) {
        const int row = row0 + r + 8 * khalf;
        if (row < n_rows)
            Y[(size_t)row * DIM + col0 + m] = c[r];
    }
}

// ---------------------------------------------------------------------------
// Zero the (poisoned) output buffer.
// ---------------------------------------------------------------------------
__global__ void zero_f4(float4* __restrict__ p, int n4) {
    const int i = blockIdx.x * blockDim.x + threadIdx.x;
    if (i < n4) { float4 z; z.x = z.y = z.z = z.w = 0.0f; p[i] = z; }
}

// ---------------------------------------------------------------------------
// out[dst[e]] += Y[src[e]]   — one wave32 per edge, float4 per lane,
// no-return global_atomic_add_f32 (STOREcnt path, L2-resident working set).
// ---------------------------------------------------------------------------
__global__ __launch_bounds__(256) void scatter_add_rows(const float* __restrict__ Y,
                                                        const int* __restrict__ src,
                                                        const int* __restrict__ dst,
                                                        float* __restrict__ out,
                                                        int n_edges) {
    const int lane = threadIdx.x & 31;
    const int wave = threadIdx.x >> 5;
    const int e    = blockIdx.x * 8 + wave;
    if (e >= n_edges) return;

    // Speculative prefetch of upcoming edge indices (global_prefetch_b8).
    __builtin_prefetch(src + e + 2048, 0, 0);
    __builtin_prefetch(dst + e + 2048, 0, 0);

    const int s = src[e];
    const int d = dst[e];
    const float4 v = *(const float4*)(Y + (size_t)s * DIM + lane * 4);
    float* o = out + (size_t)d * DIM + lane * 4;
    unsafeAtomicAdd(o + 0, v.x);
    unsafeAtomicAdd(o + 1, v.y);
    unsafeAtomicAdd(o + 2, v.z);
    unsafeAtomicAdd(o + 3, v.w);
}

// ---------------------------------------------------------------------------
// Launch
// ---------------------------------------------------------------------------
extern "C" void kernel_launch(void* const* d_in, const int* in_sizes, int n_in,
                              void* d_out, int out_size, void* d_ws, size_t ws_size,
                              hipStream_t stream) {
    const float* logits_A = (const float*)d_in[0];
    const float* logits_B = (const float*)d_in[1];
    // d_in[2], d_in[3] (frozen_A/B) do not affect the math.
    const int*   e_AB     = (const int*)d_in[4];
    const int*   e_BA     = (const int*)d_in[5];
    const int*   e_AA     = (const int*)d_in[6];
    const float* Wp_AB    = (const float*)d_in[7];
    const float* Wn_AB    = (const float*)d_in[8];
    const float* Wp_BA    = (const float*)d_in[9];
    const float* Wn_BA    = (const float*)d_in[10];
    const float* Wp_AA    = (const float*)d_in[11];
    const float* Wn_AA    = (const float*)d_in[12];

    const int NA   = in_sizes[0] / DIM;
    const int NB   = in_sizes[1] / DIM;
    const int E_AB = in_sizes[4] / 2;
    const int E_BA = in_sizes[5] / 2;
    const int E_AA = in_sizes[6] / 2;
    const int NMAX = NA > NB ? NA : NB;

    // Workspace layout (floats): 4 node-matrix buffers + 3 effective weights.
    float* ws   = (float*)d_ws;
    const size_t stride = (size_t)NMAX * DIM;
    float* pA   = ws;                  // softmax(logits_A)
    float* pB   = ws + 1 * stride;     // softmax(logits_B), later reused as Y_AA
    float* Y_AB = ws + 2 * stride;     // pA @ W_AB
    float* Y_BA = ws + 3 * stride;     // pB @ W_BA
    float* W_AB = ws + 4 * stride;
    float* W_BA = W_AB + DIM * DIM;
    float* W_AA = W_BA + DIM * DIM;
    float* Y_AA = pB;                  // pB dead after Y_BA gemm is enqueued

    float* outA = (float*)d_out;                    // delta_A [NA x 128]
    float* outB = outA + (size_t)NA * DIM;          // delta_B [NB x 128]

    // 1) Effective weights: W = W_pos - softplus(W_neg_raw)
    {
        const int n = DIM * DIM, blk = 256, grd = (n + blk - 1) / blk;
        weight_eff<<<grd, blk, 0, stream>>>(Wp_AB, Wn_AB, W_AB, n);
        weight_eff<<<grd, blk, 0, stream>>>(Wp_BA, Wn_BA, W_BA, n);
        weight_eff<<<grd, blk, 0, stream>>>(Wp_AA, Wn_AA, W_AA, n);
    }

    // 2) Row softmax for both node sets.
    softmax_rows<<<(NA + 7) / 8, 256, 0, stream>>>(logits_A, pA, NA);
    softmax_rows<<<(NB + 7) / 8, 256, 0, stream>>>(logits_B, pB, NB);

    // 3) Node-level GEMMs (linearity: segment_sum(P[src]@W) == scatter((P@W)[src])).
    gemm_wmma_f32<<<(NA + 15) / 16, 256, 0, stream>>>(pA, W_AB, Y_AB, NA);
    gemm_wmma_f32<<<(NB + 15) / 16, 256, 0, stream>>>(pB, W_BA, Y_BA, NB);
    gemm_wmma_f32<<<(NA + 15) / 16, 256, 0, stream>>>(pA, W_AA, Y_AA, NA);

    // 4) Zero output (harness poisons it), then scatter-add Y rows per edge.
    {
        const int n4 = out_size / 4, blk = 256, grd = (n4 + blk - 1) / blk;
        zero_f4<<<grd, blk, 0, stream>>>((float4*)d_out, n4);
    }
    scatter_add_rows<<<(E_BA + 7) / 8, 256, 0, stream>>>(Y_BA, e_BA, e_BA + E_BA, outA, E_BA);
    scatter_add_rows<<<(E_AA + 7) / 8, 256, 0, stream>>>(Y_AA, e_AA, e_AA + E_AA, outA, E_AA);
    scatter_add_rows<<<(E_AB + 7) / 8, 256, 0, stream>>>(Y_AB, e_AB, e_AB + E_AB, outB, E_AB);
}